// _ProposalLayer_2989297238354
// MI455X (gfx1250) — compile-verified
//
#include <hip/hip_runtime.h>
#include <stdint.h>

// ---------------- problem constants (from reference setup_inputs) ----------
#define B_      16
#define A_      9
#define H_      128
#define W_      128
#define HW_     (H_*W_)
#define K_      (HW_*A_)        // 147456 proposals per batch
#define PRE_N   6000
#define POST_N  300
#define NMS_TH  0.7f
#define NEGV    (-1e9f)
#define NBINS   65536

// Anchors for base 16, ratios {0.5,1,2}, scales {8,16,32}: all centers are
// (7.5,7.5) -> cx = 8 + 16*w, cy = 8 + 16*h; only w/h per anchor vary.
__device__ __constant__ float c_aw[9] = {184.f,368.f,736.f,128.f,256.f,512.f, 88.f,176.f,352.f};
__device__ __constant__ float c_ah[9] = { 96.f,192.f,384.f,128.f,256.f,512.f,176.f,352.f,704.f};

__device__ __forceinline__ uint32_t fkey16(float s) {
    uint32_t u = __float_as_uint(s);
    u = (u & 0x80000000u) ? ~u : (u | 0x80000000u);   // order-preserving key
    return u >> 16;
}

// ---------------- kernel 0: init scratch ----------------------------------
__global__ void k_init(uint32_t* __restrict__ hist, float4* __restrict__ sel_boxes,
                       float* __restrict__ sel_scores, uint32_t* __restrict__ counters) {
    int i = blockIdx.x * blockDim.x + threadIdx.x;
    if (i < B_ * NBINS) hist[i] = 0u;
    if (i < B_ * PRE_N) { sel_scores[i] = NEGV; sel_boxes[i] = make_float4(0.f,0.f,0.f,0.f); }
    if (i < 2 * B_)     counters[i] = 0u;
}

// ---------------- kernel 1: bbox decode + clip + score histogram ----------
__global__ void k_transform(const float* __restrict__ cls, const float* __restrict__ deltas,
                            const float* __restrict__ imsh, float4* __restrict__ boxes_all,
                            uint32_t* __restrict__ hist) {
    int t = blockIdx.x * blockDim.x + threadIdx.x;
    if (t >= B_ * K_) return;
    int b  = t / K_;
    int r  = t - b * K_;
    int hw = r / A_;
    int a  = r - hw * A_;
    int h  = hw / W_;
    int w  = hw - h * W_;

    // scores = cls_score[:,1] transposed (B,H,W,A)
    float s = cls[(((size_t)b * 2 + 1) * A_ + a) * HW_ + hw];
    atomicAdd(&hist[(size_t)b * NBINS + fkey16(s)], 1u);

    // deltas[b, a*4+c, h, w]
    size_t dbase = (((size_t)b * 4 * A_) + (size_t)a * 4) * HW_ + hw;
    float dx = deltas[dbase];
    float dy = deltas[dbase + 1 * HW_];
    float dw = deltas[dbase + 2 * HW_];
    float dh = deltas[dbase + 3 * HW_];

    float aw = c_aw[a], ah = c_ah[a];
    float cx = 8.0f + 16.0f * (float)w;
    float cy = 8.0f + 16.0f * (float)h;
    float pcx = dx * aw + cx;
    float pcy = dy * ah + cy;
    float pw  = expf(dw) * aw;
    float ph  = expf(dh) * ah;

    float wmax = imsh[b * 2 + 1] - 1.0f;
    float hmax = imsh[b * 2 + 0] - 1.0f;
    float x1 = fminf(fmaxf(pcx - 0.5f * pw, 0.f), wmax);
    float y1 = fminf(fmaxf(pcy - 0.5f * ph, 0.f), hmax);
    float x2 = fminf(fmaxf(pcx + 0.5f * pw, 0.f), wmax);
    float y2 = fminf(fmaxf(pcy + 0.5f * ph, 0.f), hmax);
    boxes_all[t] = make_float4(x1, y1, x2, y2);
}

// ---------------- kernel 2: find per-batch top-6000 threshold bin ---------
__global__ void k_thresh(const uint32_t* __restrict__ hist, uint32_t* __restrict__ thresh) {
    __shared__ uint32_t chunk[256];
    int b = blockIdx.x, tid = threadIdx.x;
    const uint32_t* hb = hist + (size_t)b * NBINS;
    uint32_t sum = 0;
    #pragma unroll 8
    for (int i = 0; i < 256; ++i) sum += hb[tid * 256 + i];
    chunk[tid] = sum;
    __syncthreads();
    if (tid == 0) {
        uint32_t acc = 0, T = 0, nab = 0;
        bool found = false;
        for (int c = 255; c >= 0 && !found; --c) {
            uint32_t cs = chunk[c];
            if (acc + cs >= PRE_N) {
                for (int bin = c * 256 + 255; bin >= c * 256; --bin) {
                    uint32_t hv = hb[bin];
                    if (acc + hv >= PRE_N) { T = (uint32_t)bin; nab = acc; found = true; break; }
                    acc += hv;
                }
            } else {
                acc += cs;
            }
        }
        if (!found) { T = 0; nab = acc; }
        thresh[b * 2 + 0] = T;
        thresh[b * 2 + 1] = nab;
    }
}

// ---------------- kernel 3: compact top-6000 candidates -------------------
__global__ void k_compact(const float* __restrict__ cls, const float4* __restrict__ boxes_all,
                          const uint32_t* __restrict__ thresh, uint32_t* __restrict__ counters,
                          float4* __restrict__ sel_boxes, float* __restrict__ sel_scores) {
    int t = blockIdx.x * blockDim.x + threadIdx.x;
    if (t >= B_ * K_) return;
    int b  = t / K_;
    int r  = t - b * K_;
    int hw = r / A_;
    int a  = r - hw * A_;
    float s = cls[(((size_t)b * 2 + 1) * A_ + a) * HW_ + hw];
    uint32_t k16 = fkey16(s);
    uint32_t T   = thresh[b * 2 + 0];
    uint32_t nab = thresh[b * 2 + 1];
    if (k16 < T) return;
    uint32_t pos;
    if (k16 > T) {
        pos = atomicAdd(&counters[b * 2 + 0], 1u);      // guaranteed < PRE_N
    } else {
        pos = nab + atomicAdd(&counters[b * 2 + 1], 1u);
        if (pos >= PRE_N) return;                       // boundary-bin overflow
    }
    sel_boxes [(size_t)b * PRE_N + pos] = boxes_all[t];
    sel_scores[(size_t)b * PRE_N + pos] = s;
}

// ---------------- TDM: 1-D tensor_load_to_lds (CDNA5, data_size = 4B) -----
typedef unsigned int u32v4 __attribute__((ext_vector_type(4)));
typedef int          i32v8 __attribute__((ext_vector_type(8)));
typedef int          i32v4 __attribute__((ext_vector_type(4)));

__device__ __forceinline__ void tdm_load_1d(uint32_t lds_off, const void* gsrc, uint32_t elems) {
    uint64_t ga = (uint64_t)(uintptr_t)gsrc;
    u32v4 g0;
    g0[0] = 1u;                                             // count=1, user D#
    g0[1] = lds_off;                                        // lds_addr (bytes)
    g0[2] = (uint32_t)ga;                                   // global_addr[31:0]
    g0[3] = (uint32_t)((ga >> 32) & 0x01FFFFFFu) | (2u << 30); // addr[56:32] | type=2
    i32v8 g1;
    g1[0] = (int)(2u << 16);                                // data_size=2 (4 bytes)
    g1[1] = (int)((elems & 0xFFFFu) << 16);                 // tensor_dim0[15:0]
    g1[2] = (int)((elems >> 16) | (1u << 16));              // tensor_dim0[31:16] | tensor_dim1=1
    g1[3] = (int)((elems & 0xFFFFu) << 16);                 // tile_dim0 = elems (<=65535)
    g1[4] = 0;                                              // tile_dim1=0, tile_dim2=0
    g1[5] = (int)elems;                                     // tensor_dim0_stride[31:0]
    g1[6] = 0;
    g1[7] = 0;
    i32v4 gz4; gz4[0] = gz4[1] = gz4[2] = gz4[3] = 0;       // groups 2/3 unused (<=2D)
    i32v8 gz8;
    gz8[0] = gz8[1] = gz8[2] = gz8[3] = 0;
    gz8[4] = gz8[5] = gz8[6] = gz8[7] = 0;
    // clang-23 / therock-10.0 form: 6 args (g0, g1, g2, g3, extra, cpol)
    __builtin_amdgcn_tensor_load_to_lds(g0, g1, gz4, gz4, gz8, 0);
}

// ---------------- kernel 4: LDS-resident greedy NMS (1 WG per batch) ------
__global__ void __launch_bounds__(1024) k_nms(const float4* __restrict__ sel_boxes,
                                              const float*  __restrict__ sel_scores,
                                              float* __restrict__ out) {
    __shared__ float4 s_box[PRE_N];     // 96000 B
    __shared__ float  s_sc [PRE_N];     // 24000 B  (live scores, mutated)
    __shared__ float  s_ar [PRE_N];     // 24000 B  (areas)
    __shared__ float  s_rs[32];
    __shared__ int    s_ri[32];
    __shared__ float4 s_pb;
    __shared__ float  s_pa;
    __shared__ int    s_pi;
    __shared__ int    s_pv;

    const int b = blockIdx.x, tid = threadIdx.x;

    // Bulk global->LDS via Tensor Data Mover, issued once by wave 0.
    if (tid < 32) {
        uint32_t off_box = (uint32_t)(uintptr_t)&s_box[0];  // low 32 bits = LDS offset
        uint32_t off_sc  = (uint32_t)(uintptr_t)&s_sc[0];
        tdm_load_1d(off_box, sel_boxes  + (size_t)b * PRE_N, PRE_N * 4u);
        tdm_load_1d(off_sc,  sel_scores + (size_t)b * PRE_N, PRE_N);
        __builtin_amdgcn_s_wait_tensorcnt((short)0);
    }
    __syncthreads();

    for (int j = tid; j < PRE_N; j += 1024) {
        float4 c = s_box[j];
        s_ar[j] = (c.z - c.x + 1.0f) * (c.w - c.y + 1.0f);
    }
    __syncthreads();

    for (int it = 0; it < POST_N; ++it) {
        // block-wide argmax over live scores (first occurrence on ties)
        float best = -INFINITY;
        int   bi   = 0x7FFFFFFF;
        for (int j = tid; j < PRE_N; j += 1024) {
            float s = s_sc[j];
            if (s > best) { best = s; bi = j; }
        }
        #pragma unroll
        for (int off = 16; off > 0; off >>= 1) {
            float os = __shfl_down(best, off, 32);
            int   oi = __shfl_down(bi,   off, 32);
            if (os > best || (os == best && oi < bi)) { best = os; bi = oi; }
        }
        const int lane = tid & 31, wid = tid >> 5;
        if (lane == 0) { s_rs[wid] = best; s_ri[wid] = bi; }
        __syncthreads();
        if (wid == 0) {
            best = s_rs[lane]; bi = s_ri[lane];
            #pragma unroll
            for (int off = 16; off > 0; off >>= 1) {
                float os = __shfl_down(best, off, 32);
                int   oi = __shfl_down(bi,   off, 32);
                if (os > best || (os == best && oi < bi)) { best = os; bi = oi; }
            }
            if (lane == 0) {
                int valid = (best > NEGV * 0.5f) ? 1 : 0;
                float4 bb = s_box[bi];
                s_pi = bi; s_pv = valid; s_pb = bb; s_pa = s_ar[bi];
                float* orow = out + ((size_t)b * POST_N + it) * 5;
                orow[0] = (float)b;                         // bcol always set
                orow[1] = valid ? bb.x : 0.0f;
                orow[2] = valid ? bb.y : 0.0f;
                orow[3] = valid ? bb.z : 0.0f;
                orow[4] = valid ? bb.w : 0.0f;
            }
        }
        __syncthreads();
        if (s_pv) {
            const float4 bb = s_pb;
            const float  ai = s_pa;
            for (int j = tid; j < PRE_N; j += 1024) {
                float4 c = s_box[j];
                float xx1 = fmaxf(bb.x, c.x), yy1 = fmaxf(bb.y, c.y);
                float xx2 = fminf(bb.z, c.z), yy2 = fminf(bb.w, c.w);
                float inter = fmaxf(xx2 - xx1 + 1.0f, 0.f) * fmaxf(yy2 - yy1 + 1.0f, 0.f);
                float iou = inter / (ai + s_ar[j] - inter);
                if (iou > NMS_TH) s_sc[j] = NEGV;
            }
            if (tid == 0) s_sc[s_pi] = NEGV;                // suppress self
        }
        __syncthreads();
    }
}

// ---------------- host launcher -------------------------------------------
extern "C" void kernel_launch(void* const* d_in, const int* in_sizes, int n_in,
                              void* d_out, int out_size, void* d_ws, size_t ws_size,
                              hipStream_t stream) {
    (void)in_sizes; (void)n_in; (void)out_size; (void)ws_size;
    const float* cls    = (const float*)d_in[0];   // (16,2,9,128,128) f32
    const float* deltas = (const float*)d_in[1];   // (16,36,128,128)  f32
    const float* imsh   = (const float*)d_in[2];   // (16,2)           f32
    float* out = (float*)d_out;                    // (16,300,5)       f32

    // workspace layout (~43.9 MB)
    char* ws = (char*)d_ws;
    size_t off = 0;
    float4*   boxes_all  = (float4*)(ws + off);   off += (size_t)B_ * K_ * 16;      // 37.75 MB
    uint32_t* hist       = (uint32_t*)(ws + off); off += (size_t)B_ * NBINS * 4;    //  4.19 MB
    uint32_t* thresh     = (uint32_t*)(ws + off); off += 256;
    uint32_t* counters   = (uint32_t*)(ws + off); off += 256;
    float4*   sel_boxes  = (float4*)(ws + off);   off += (size_t)B_ * PRE_N * 16;   //  1.54 MB
    float*    sel_scores = (float*)(ws + off);    off += (size_t)B_ * PRE_N * 4;    //  0.38 MB

    k_init     <<<(B_ * NBINS + 255) / 256, 256, 0, stream>>>(hist, sel_boxes, sel_scores, counters);
    k_transform<<<(B_ * K_   + 255) / 256, 256, 0, stream>>>(cls, deltas, imsh, boxes_all, hist);
    k_thresh   <<<B_, 256, 0, stream>>>(hist, thresh);
    k_compact  <<<(B_ * K_   + 255) / 256, 256, 0, stream>>>(cls, boxes_all, thresh, counters,
                                                             sel_boxes, sel_scores);
    k_nms      <<<B_, 1024, 0, stream>>>(sel_boxes, sel_scores, out);
}